// GraphConv_79216376807728
// MI455X (gfx1250) — compile-verified
//
#include <hip/hip_runtime.h>
#include <hip/hip_bf16.h>

typedef __attribute__((ext_vector_type(2))) float v2f;
typedef __attribute__((ext_vector_type(8))) float v8f;

#define DD    64
#define NREL  20
#define NRELP 32   // relation rows padded to two full WMMA N-tiles

// ---------------- helpers: order-preserving float<->uint key for atomicMax ---
__device__ __forceinline__ unsigned f2key(float f) {
    unsigned b = __float_as_uint(f);
    return (b & 0x80000000u) ? ~b : (b | 0x80000000u);
}
__device__ __forceinline__ float key2f(unsigned k) {
    unsigned b = (k & 0x80000000u) ? (k & 0x7FFFFFFFu) : ~k;
    return __uint_as_float(b);
}

// ---------------- init: res = entity_emb, segkey/segsum = 0 -------------------
__global__ __launch_bounds__(256)
void init_kernel(const float* __restrict__ ent, float* __restrict__ res,
                 unsigned* __restrict__ segkey, float* __restrict__ segsum,
                 int N, long NE)
{
    long i = (long)blockIdx.x * 256 + threadIdx.x;
    if (i < NE) res[i] = ent[i];
    if (i < N) { segkey[i] = 0u; segsum[i] = 0.0f; }
}

__global__ __launch_bounds__(256)
void zero_kernel(float* __restrict__ p, long n)
{
    long i = (long)blockIdx.x * 256 + threadIdx.x;
    if (i < n) p[i] = 0.0f;
}

// ---------------- edge scores via fp32 WMMA: kg[e] = (U @ R^T)[e, type(e)] ----
// 16 edges per wave. A tile: u[m,d] = ent[head[m],d]*ent[tail[m],d].
// Relation matrix zero-padded to 32 rows in LDS so both B tiles load
// unconditionally (keeps EXEC uniform, no saveexec in the K-loop).
__global__ __launch_bounds__(256)
void score_wmma_kernel(const float* __restrict__ ent, const float* __restrict__ relm,
                       const int* __restrict__ head, const int* __restrict__ tail,
                       const int* __restrict__ etype, float* __restrict__ kg, long E)
{
    __shared__ float rel_s[NRELP][DD];    // 8 KB: relation matrix, rows 20..31 zero
    __shared__ float sc[8][2][16][17];    // per-wave D-tile spill for extraction

    const int tid = threadIdx.x;
    for (int i = tid; i < NRELP * DD; i += 256) {
        const int r = i / DD, c = i % DD;
        rel_s[r][c] = (r < NREL) ? relm[r * DD + c] : 0.0f;
    }
    __syncthreads();

    const int wave = tid >> 5;
    const int lane = tid & 31;
    const int lo   = lane & 15;   // A: row m / B: col n / D: col n
    const int hi   = lane >> 4;   // K sub-pair selector

    const long base = (long)blockIdx.x * 128 + (long)wave * 16;
    const long em   = base + lo;                 // edge owned by this lane pair
    const long emc  = (em < E) ? em : (E - 1);   // clamped for safe loads
    const int  h = head[emc];
    const int  t = tail[emc];
    const float* __restrict__ hrow = ent + (long)h * DD;
    const float* __restrict__ trow = ent + (long)t * DD;

    v8f acc0 = {};   // relations 0..15
    v8f acc1 = {};   // relations 16..19 (rows 20..31 are zero-padded)

    #pragma unroll
    for (int k = 0; k < 16; ++k) {
        const int d = 4 * k + 2 * hi;
        v2f a, b0, b1;
        a.x  = hrow[d]     * trow[d];
        a.y  = hrow[d + 1] * trow[d + 1];
        b0.x = rel_s[lo][d];
        b0.y = rel_s[lo][d + 1];
        b1.x = rel_s[16 + lo][d];
        b1.y = rel_s[16 + lo][d + 1];
        acc0 = __builtin_amdgcn_wmma_f32_16x16x4_f32(false, a, false, b0,
                                                     (short)0, acc0, false, false);
        acc1 = __builtin_amdgcn_wmma_f32_16x16x4_f32(false, a, false, b1,
                                                     (short)0, acc1, false, false);
    }

    // D layout: lane (lo, hi) holds D[m = v + 8*hi, n = lo] in acc[v]
    #pragma unroll
    for (int v = 0; v < 8; ++v) {
        const int mm = v + 8 * hi;
        sc[wave][0][mm][lo] = acc0[v];
        sc[wave][1][mm][lo] = acc1[v];
    }
    __syncthreads();

    if (lane < 16 && em < E) {
        const int ty = etype[em] - 1;          // 0..19
        const float s = (ty < 16) ? sc[wave][0][lane][ty]
                                  : sc[wave][1][lane][ty - 16];
        kg[em] = s;
    }
}

// ---------------- segment softmax over head ----------------------------------
__global__ __launch_bounds__(256)
void seg_max_kernel(const float* __restrict__ kg, const int* __restrict__ head,
                    unsigned* __restrict__ segkey, long E)
{
    long e = (long)blockIdx.x * 256 + threadIdx.x;
    if (e >= E) return;
    atomicMax(&segkey[head[e]], f2key(kg[e]));
}

__global__ __launch_bounds__(256)
void seg_exp_kernel(float* __restrict__ kg, const int* __restrict__ head,
                    const unsigned* __restrict__ segkey, float* __restrict__ segsum,
                    long E)
{
    long e = (long)blockIdx.x * 256 + threadIdx.x;
    if (e >= E) return;
    const int h = head[e];
    const float ex = __expf(kg[e] - key2f(segkey[h]));
    kg[e] = ex;
    atomicAdd(&segsum[h], ex);
}

__global__ __launch_bounds__(256)
void seg_div_kernel(float* __restrict__ kg, const int* __restrict__ head,
                    const float* __restrict__ segsum, long E)
{
    long e = (long)blockIdx.x * 256 + threadIdx.x;
    if (e >= E) return;
    kg[e] = kg[e] / segsum[head[e]];
}

// ---------------- hop: agg[head] += score * (rel o x[tail]) -------------------
// 16 lanes per edge, float4 per lane -> 64 floats; 128-bit coalesced gathers.
__global__ __launch_bounds__(256)
void hop_scatter_kernel(const float* __restrict__ xin, const float* __restrict__ relm,
                        const int* __restrict__ head, const int* __restrict__ tail,
                        const int* __restrict__ etype, const float* __restrict__ score,
                        float* __restrict__ agg, long E)
{
    long gid = (long)blockIdx.x * 256 + threadIdx.x;
    long e = gid >> 4;
    if (e >= E) return;
    const int d = (int)(gid & 15) * 4;
    const int h = head[e];
    const int t = tail[e];
    const int r = etype[e] - 1;
    const float s = score[e];
    const float4 xr = *(const float4*)(xin  + (long)t * DD + d);
    const float4 rr = *(const float4*)(relm + (long)r * DD + d);
    float* o = agg + (long)h * DD + d;
    atomicAdd(o + 0, s * rr.x * xr.x);
    atomicAdd(o + 1, s * rr.y * xr.y);
    atomicAdd(o + 2, s * rr.z * xr.z);
    atomicAdd(o + 3, s * rr.w * xr.w);
}

// ---------------- hop: x = agg / max(||agg||, eps); res += x ------------------
// one wave32 per entity row: lane handles d=lane and d=lane+32
__global__ __launch_bounds__(256)
void hop_norm_kernel(const float* __restrict__ agg, float* __restrict__ xout,
                     float* __restrict__ res, int N)
{
    const int wave = threadIdx.x >> 5;
    const int lane = threadIdx.x & 31;
    const long row = (long)blockIdx.x * 8 + wave;
    if (row >= N) return;
    const float* a = agg + row * DD;
    const float v0 = a[lane];
    const float v1 = a[lane + 32];
    float ss = v0 * v0 + v1 * v1;
    #pragma unroll
    for (int off = 16; off > 0; off >>= 1)
        ss += __shfl_xor(ss, off, 32);
    const float inv = 1.0f / fmaxf(sqrtf(ss), 1e-12f);
    const float x0 = v0 * inv, x1 = v1 * inv;
    xout[row * DD + lane]      = x0;
    xout[row * DD + lane + 32] = x1;
    res[row * DD + lane]      += x0;
    res[row * DD + lane + 32] += x1;
}

// -----------------------------------------------------------------------------
extern "C" void kernel_launch(void* const* d_in, const int* in_sizes, int n_in,
                              void* d_out, int out_size, void* d_ws, size_t ws_size,
                              hipStream_t stream)
{
    const float* ent   = (const float*)d_in[0];
    const float* relm  = (const float*)d_in[1];
    const int*   eidx  = (const int*)d_in[2];
    const int*   etype = (const int*)d_in[3];

    const long E = in_sizes[3];
    const int  N = in_sizes[0] / DD;
    const int* head = eidx;
    const int* tail = eidx + E;

    char* ws = (char*)d_ws;
    auto al = [](size_t x) { return (x + 255) & ~(size_t)255; };
    size_t off = 0;
    float*    score  = (float*)(ws + off);    off = al(off + (size_t)E * 4);
    unsigned* segkey = (unsigned*)(ws + off); off = al(off + (size_t)N * 4);
    float*    segsum = (float*)(ws + off);    off = al(off + (size_t)N * 4);
    float*    agg    = (float*)(ws + off);    off = al(off + (size_t)N * DD * 4);
    float*    xbuf   = (float*)(ws + off);    off = al(off + (size_t)N * DD * 4);
    float*    res    = (float*)d_out;

    const long NE = (long)N * DD;

    init_kernel<<<(int)((NE + 255) / 256), 256, 0, stream>>>(ent, res, segkey, segsum, N, NE);
    score_wmma_kernel<<<(int)((E + 127) / 128), 256, 0, stream>>>(ent, relm, head, tail,
                                                                  etype, score, E);
    seg_max_kernel<<<(int)((E + 255) / 256), 256, 0, stream>>>(score, head, segkey, E);
    seg_exp_kernel<<<(int)((E + 255) / 256), 256, 0, stream>>>(score, head, segkey, segsum, E);
    seg_div_kernel<<<(int)((E + 255) / 256), 256, 0, stream>>>(score, head, segsum, E);

    const float* xin = ent;
    for (int hop = 0; hop < 3; ++hop) {
        zero_kernel<<<(int)((NE + 255) / 256), 256, 0, stream>>>(agg, NE);
        hop_scatter_kernel<<<(int)((E * 16 + 255) / 256), 256, 0, stream>>>(
            xin, relm, head, tail, etype, score, agg, E);
        hop_norm_kernel<<<(N + 7) / 8, 256, 0, stream>>>(agg, xbuf, res, N);
        xin = xbuf;
    }
}